// TripletAttention_64750926954822
// MI455X (gfx1250) — compile-verified
//
#include <hip/hip_runtime.h>
#include <stdint.h>

typedef __attribute__((ext_vector_type(2))) float v2f;
typedef __attribute__((ext_vector_type(4))) float f4;
typedef __attribute__((ext_vector_type(8))) float v8f;
typedef __attribute__((ext_vector_type(4))) unsigned int u32x4;
typedef __attribute__((ext_vector_type(8))) int i32x8;
typedef __attribute__((ext_vector_type(4))) int i32x4;

constexpr int N = 32, C = 256, H = 56, W = 56;
constexpr float BN_EPS = 1e-5f;
constexpr float NEG_BIG = -3.402823466e38f;

__device__ __forceinline__ float sigmoidf(float v) { return 1.0f / (1.0f + __expf(-v)); }

// ---------------- reductions (HBM-streaming, bandwidth bound) ----------------
// comp_s[n][2][h][w]: max/mean over C
__global__ void reduce_c_kernel(const float* __restrict__ x, float* __restrict__ comp_s) {
  int id = blockIdx.x * blockDim.x + threadIdx.x;
  if (id >= N * H * W) return;
  int w = id % W; int t = id / W; int h = t % H; int n = t / H;
  const float* p = x + ((size_t)n * C * H + h) * W + w;
  float mx = NEG_BIG, sm = 0.f;
  for (int c = 0; c < C; ++c) {
    float v = p[(size_t)c * H * W];
    mx = fmaxf(mx, v); sm += v;
  }
  size_t o = ((size_t)(n * 2 + 0) * H + h) * W + w;
  comp_s[o] = mx;
  comp_s[o + (size_t)H * W] = sm * (1.0f / C);
}

// comp_h[n][2][c][w]: max/mean over H
__global__ void reduce_h_kernel(const float* __restrict__ x, float* __restrict__ comp_h) {
  int id = blockIdx.x * blockDim.x + threadIdx.x;
  if (id >= N * C * W) return;
  int w = id % W; int t = id / W; int c = t % C; int n = t / C;
  const float* p = x + ((size_t)(n * C + c) * H) * W + w;
  float mx = NEG_BIG, sm = 0.f;
  for (int h = 0; h < H; ++h) {
    float v = p[(size_t)h * W];
    mx = fmaxf(mx, v); sm += v;
  }
  size_t o = ((size_t)(n * 2 + 0) * C + c) * W + w;
  comp_h[o] = mx;
  comp_h[o + (size_t)C * W] = sm * (1.0f / H);
}

// comp_w[n][2][h][c]: max/mean over W — one wave per (n,h,c), coalesced row read
__global__ void reduce_w_kernel(const float* __restrict__ x, float* __restrict__ comp_w) {
  int lane = threadIdx.x & 31;
  int wv = blockIdx.x * (blockDim.x >> 5) + (threadIdx.x >> 5);
  if (wv >= N * H * C) return;
  int c = wv % C; int t = wv / C; int h = t % H; int n = t / H;
  const float* row = x + ((size_t)(n * C + c) * H + h) * W;
  float v1 = row[lane];                                  // lane < 32 < W
  float v2 = (lane + 32 < W) ? row[lane + 32] : NEG_BIG; // lanes 0..23
  float mx = fmaxf(v1, v2);
  float sm = v1 + ((lane + 32 < W) ? v2 : 0.f);
  for (int off = 16; off >= 1; off >>= 1) {
    mx = fmaxf(mx, __shfl_xor(mx, off, 32));
    sm += __shfl_xor(sm, off, 32);
  }
  if (lane == 0) {
    size_t o = ((size_t)(n * 2 + 0) * H + h) * C + c;
    comp_w[o] = mx;
    comp_w[o + (size_t)H * C] = sm * (1.0f / W);
  }
}

// y[n][c] = mean over (H,W) = mean over h of comp_w mean-channel
__global__ void reduce_y_kernel(const float* __restrict__ comp_w, float* __restrict__ y) {
  int id = blockIdx.x * blockDim.x + threadIdx.x;
  if (id >= N * C) return;
  int c = id % C; int n = id / C;
  const float* p = comp_w + ((size_t)(n * 2 + 1) * H) * C + c;
  float sm = 0.f;
  for (int h = 0; h < H; ++h) sm += p[(size_t)h * C];
  y[id] = sm * (1.0f / H);
}

// ---------------- 7x7 conv + BN + sigmoid via WMMA f32 16x16x4 ----------------
// in: comp[n][2][D1][D2], weights wk[2*7*7], out gate[n][D1][D2]
// One wave computes 16 outputs along D2 at fixed (n,d1).
// Stage the 2x7x22 halo patch into per-wave LDS via async global->LDS copies,
// then im2col GEMV on WMMA: A[m,k] from LDS, B[k,*] = wk[k] replicated columns.
constexpr int TPITCH = 24;              // 22 used cols, padded
constexpr int TELEMS = 2 * 7 * TPITCH;  // 336 floats / wave

__global__ void conv_gate_kernel(const float* __restrict__ comp, const float* __restrict__ wk,
                                 const float* __restrict__ gp, const float* __restrict__ bp,
                                 const float* __restrict__ mp, const float* __restrict__ vp,
                                 float* __restrict__ gate, int D1, int D2) {
  __shared__ float tile[4][TELEMS];
  int lane = threadIdx.x & 31;
  int wid = threadIdx.x >> 5;
  int wv = blockIdx.x * (blockDim.x >> 5) + wid;
  int tiles = (D2 + 15) >> 4;
  int td2 = wv % tiles; int t = wv / tiles;
  int d1 = t % D1; int n = t / D1;
  if (n >= N) return;                         // uniform per wave
  int d2b = td2 << 4;
  int mrow = lane & 15;
  int khalf = (lane & 16) ? 2 : 0;            // A 16x4: lanes 0-15 -> K0,K1 ; 16-31 -> K2,K3

  // ---- stage patch into LDS: zero-fill halo, async-load valid elements ----
  float* buf = tile[wid];
  for (int e = lane; e < TELEMS; e += 32) buf[e] = 0.f;
  asm volatile("s_wait_dscnt 0x0" ::: "memory");
  unsigned ldsbase = (unsigned)(uintptr_t)buf;
  for (int e = lane; e < TELEMS; e += 32) {
    int col = e % TPITCH; int rr = e / TPITCH;
    int r = rr % 7; int ch = rr / 7;
    int e1 = d1 + r - 3;
    int e2 = d2b + col - 3;
    if (col < 22 && e1 >= 0 && e1 < D1 && e2 >= 0 && e2 < D2) {
      unsigned long long ga =
          (unsigned long long)(uintptr_t)(comp + ((size_t)(n * 2 + ch) * D1 + e1) * D2 + e2);
      unsigned ldsoff = ldsbase + (unsigned)e * 4u;
      asm volatile("global_load_async_to_lds_b32 %0, %1, off"
                   :: "v"(ldsoff), "v"(ga) : "memory");
    }
  }
  asm volatile("s_wait_asynccnt 0x0" ::: "memory");

  // ---- fully unrolled K loop: 25 x wmma 16x16x4 (K=98 padded to 100) ----
  v8f acc = {0.f, 0.f, 0.f, 0.f, 0.f, 0.f, 0.f, 0.f};
#pragma unroll
  for (int k0 = 0; k0 < 100; k0 += 4) {
    v2f a, b;
#pragma unroll
    for (int tt = 0; tt < 2; ++tt) {
      int k = k0 + khalf + tt;                // compile-time k0 -> decode folds to selects
      float av = 0.f, bv = 0.f;
      if (k < 98) {
        bv = wk[k];
        int ch = k / 49; int rem = k - ch * 49;
        int r = rem / 7;  int s = rem - r * 7;
        av = buf[(ch * 7 + r) * TPITCH + mrow + s];
      }
      a[tt] = av; b[tt] = bv;
    }
    acc = __builtin_amdgcn_wmma_f32_16x16x4_f32(false, a, false, b, (short)0, acc,
                                                false, false);
  }

  // All B columns equal -> lane 0 holds o[0..7] in acc[0..7], lane 16 holds o[8..15].
  float scale = rsqrtf(vp[0] + BN_EPS);
  float gg = gp[0], bb = bp[0], mm = mp[0];
  if (lane == 0 || lane == 16) {
    int mbase = (lane == 16) ? 8 : 0;
#pragma unroll
    for (int r = 0; r < 8; ++r) {
      int d2 = d2b + mbase + r;
      if (d2 < D2) {
        float o = (acc[r] - mm) * scale * gg + bb;
        gate[((size_t)n * D1 + d1) * D2 + d2] = sigmoidf(o);
      }
    }
  }
}

// ---------------- ECA with sort/unsort (bitonic, 256 ch per image) ----------------
// The contiguous 256-float channel row is pulled into LDS by the Tensor Data Mover.
__global__ void eca_kernel(const float* __restrict__ y, const float* __restrict__ we,
                           float* __restrict__ ge) {
  __shared__ float key[256];
  __shared__ int   sidx[256];
  int n = blockIdx.x; int t = threadIdx.x;

  if (threadIdx.x < 32) {  // one wave issues the TDM load of y[n, 0:256] -> key[]
    unsigned long long ga = (unsigned long long)(uintptr_t)(y + (size_t)n * C);
    unsigned ldsaddr = (unsigned)(uintptr_t)&key[0];
    u32x4 g0;
    g0[0] = 1u;                                            // count=1, user descriptor
    g0[1] = ldsaddr;                                       // lds_addr
    g0[2] = (unsigned)ga;                                  // global_addr[31:0]
    g0[3] = ((unsigned)(ga >> 32) & 0x01FFFFFFu) | 0x80000000u;  // addr[56:32] | type=2
    i32x8 g1;
    g1[0] = 2 << 16;          // data_size=4B, no multicast/pad/iterate
    g1[1] = 256 << 16;        // tensor_dim0 = 256 (low 16 bits into bits 63:48)
    g1[2] = 1 << 16;          // tensor_dim0 hi = 0 ; tensor_dim1 = 1
    g1[3] = 256 << 16;        // tensor_dim1 hi = 0 ; tile_dim0 = 256
    g1[4] = 1;                // tile_dim1 = 1 ; tile_dim2 = 0
    g1[5] = 256;              // tensor_dim0_stride = 256
    g1[6] = 0;
    g1[7] = 0;
    i32x4 z4 = {0, 0, 0, 0};
    i32x8 z8 = {0, 0, 0, 0, 0, 0, 0, 0};
    __builtin_amdgcn_tensor_load_to_lds(g0, g1, z4, z4, z8, 0);
    __builtin_amdgcn_s_wait_tensorcnt(0);
  }
  sidx[t] = t;
  __syncthreads();

  // bitonic sort, descending
  for (int k = 2; k <= 256; k <<= 1) {
    for (int j = k >> 1; j > 0; j >>= 1) {
      int ixj = t ^ j;
      if (ixj > t) {
        float a = key[t], b = key[ixj];
        bool desc = ((t & k) == 0);
        bool sw = desc ? (a < b) : (a > b);
        if (sw) {
          key[t] = b; key[ixj] = a;
          int tmp = sidx[t]; sidx[t] = sidx[ixj]; sidx[ixj] = tmp;
        }
      }
      __syncthreads();
    }
  }
  // 1D conv k=5 pad=2 over sorted sequence, sigmoid, scatter back to channel order
  float o = 0.f;
#pragma unroll
  for (int j = 0; j < 5; ++j) {
    int p = t + j - 2;
    if (p >= 0 && p < 256) o += key[p] * we[j];
  }
  ge[n * C + sidx[t]] = sigmoidf(o);
}

// ---------------- final fused gating pass (streaming, NT hints) ----------------
__global__ void final_kernel(const float* __restrict__ x, const float* __restrict__ s_s,
                             const float* __restrict__ s_h, const float* __restrict__ s_w,
                             const float* __restrict__ ge, float* __restrict__ out) {
  size_t i4 = (size_t)blockIdx.x * blockDim.x + threadIdx.x;
  size_t total4 = (size_t)N * C * H * W / 4;
  if (i4 >= total4) return;
  size_t base = i4 * 4;
  int w = (int)(base % W); size_t t1 = base / W;
  int h = (int)(t1 % H);   size_t t2 = t1 / H;
  int c = (int)(t2 % C);   int n = (int)(t2 / C);

  f4 xv = __builtin_nontemporal_load((const f4*)(x + base));
  f4 gs = *(const f4*)(s_s + ((size_t)n * H + h) * W + w);
  f4 gh = *(const f4*)(s_h + ((size_t)n * C + c) * W + w);
  float gw = s_w[((size_t)n * H + h) * C + c];
  float gc = ge[n * C + c];

  f4 gate = 0.23779f * gs + 0.24695f * gh + (0.24365f * gw + 0.27173f * gc);
  __builtin_nontemporal_store(xv * gate, (f4*)(out + base));
}

// ---------------- launch ----------------
extern "C" void kernel_launch(void* const* d_in, const int* in_sizes, int n_in,
                              void* d_out, int out_size, void* d_ws, size_t ws_size,
                              hipStream_t stream) {
  const float* x     = (const float*)d_in[0];
  // d_in[1] = 'weight' — unused by the reference
  const float* w_h   = (const float*)d_in[2];
  const float* g_h   = (const float*)d_in[3];
  const float* b_h   = (const float*)d_in[4];
  const float* m_h   = (const float*)d_in[5];
  const float* v_h   = (const float*)d_in[6];
  const float* w_w   = (const float*)d_in[7];
  const float* g_w   = (const float*)d_in[8];
  const float* b_w   = (const float*)d_in[9];
  const float* m_w   = (const float*)d_in[10];
  const float* v_w   = (const float*)d_in[11];
  const float* w_s   = (const float*)d_in[12];
  const float* g_s   = (const float*)d_in[13];
  const float* b_s   = (const float*)d_in[14];
  const float* m_s   = (const float*)d_in[15];
  const float* v_s   = (const float*)d_in[16];
  const float* w_eca = (const float*)d_in[17];
  float* out = (float*)d_out;
  float* ws  = (float*)d_ws;

  // workspace layout (floats); all offsets 16B aligned
  constexpr size_t OFF_COMP_S = 0;                          // [N,2,H,W] 200704
  constexpr size_t OFF_COMP_H = 200704;                     // [N,2,C,W] 917504
  constexpr size_t OFF_COMP_W = OFF_COMP_H + 917504;        // [N,2,H,C] 917504
  constexpr size_t OFF_Y      = OFF_COMP_W + 917504;        // [N,C]     8192
  constexpr size_t OFF_S_S    = OFF_Y + 8192;               // [N,H,W]   100352
  constexpr size_t OFF_S_H    = OFF_S_S + 100352;           // [N,C,W]   458752
  constexpr size_t OFF_S_W    = OFF_S_H + 458752;           // [N,H,C]   458752
  constexpr size_t OFF_G_E    = OFF_S_W + 458752;           // [N,C]     8192

  float* comp_s = ws + OFF_COMP_S;
  float* comp_h = ws + OFF_COMP_H;
  float* comp_w = ws + OFF_COMP_W;
  float* yv     = ws + OFF_Y;
  float* s_s    = ws + OFF_S_S;
  float* s_h    = ws + OFF_S_H;
  float* s_w    = ws + OFF_S_W;
  float* g_e    = ws + OFF_G_E;

  // 1) reductions over x (3 streaming passes)
  reduce_c_kernel<<<(N * H * W + 255) / 256, 256, 0, stream>>>(x, comp_s);
  reduce_h_kernel<<<(N * C * W + 255) / 256, 256, 0, stream>>>(x, comp_h);
  reduce_w_kernel<<<(N * H * C + 7) / 8, 256, 0, stream>>>(x, comp_w);
  reduce_y_kernel<<<(N * C + 255) / 256, 256, 0, stream>>>(comp_w, yv);

  // 2) 7x7 conv + BN + sigmoid gates (async->LDS staging + WMMA)
  {
    int D1 = H, D2 = W;                                   // branch 4: (H,W)
    int waves = N * D1 * ((D2 + 15) / 16);
    conv_gate_kernel<<<(waves + 3) / 4, 128, 0, stream>>>(comp_s, w_s, g_s, b_s, m_s, v_s, s_s, D1, D2);
  }
  {
    int D1 = C, D2 = W;                                   // branch 1: (C,W)
    int waves = N * D1 * ((D2 + 15) / 16);
    conv_gate_kernel<<<(waves + 3) / 4, 128, 0, stream>>>(comp_h, w_h, g_h, b_h, m_h, v_h, s_h, D1, D2);
  }
  {
    int D1 = H, D2 = C;                                   // branch 2: (H,C)
    int waves = N * D1 * ((D2 + 15) / 16);
    conv_gate_kernel<<<(waves + 3) / 4, 128, 0, stream>>>(comp_w, w_w, g_w, b_w, m_w, v_w, s_w, D1, D2);
  }

  // 3) ECA sort/conv/unsort gate (TDM row load)
  eca_kernel<<<N, 256, 0, stream>>>(yv, w_eca, g_e);

  // 4) fused output pass
  {
    size_t total4 = (size_t)N * C * H * W / 4;
    final_kernel<<<(unsigned)((total4 + 255) / 256), 256, 0, stream>>>(x, s_s, s_h, s_w, g_e, out);
  }
}